// QLSTM_65481071410483
// MI455X (gfx1250) — compile-verified
//
#include <hip/hip_runtime.h>
#include <hip/hip_bf16.h>
#include <math.h>

#define TT 512
#define BB 1024
#define DIN 64
#define NH 8

typedef __attribute__((ext_vector_type(2))) float v2f;
typedef __attribute__((ext_vector_type(8))) float v8f;

// ---------------------------------------------------------------------------
// Pack weights directly into WMMA B-fragment layout.
// Logical B = [Wf_x | Wi_x | Wu_x | Wo_x | Wi2k | 0pad] : 64 x 48.
// Fragment layout for V_WMMA_F32_16X16X4_F32 (B is 4x16 per op):
//   lanes 0-15  hold rows K=k0,k0+1 (VGPR0,VGPR1) at N=lane
//   lanes 16-31 hold rows K=k0+2,k0+3           at N=lane-16
// Bfrag[((kstep*3 + ntile)*32 + lane)] = float2( B[krow][col], B[krow+1][col] )
//   with krow = kstep*4 + (lane>>4)*2, col = ntile*16 + (lane&15).
// Also packs biases into bpack[48].
// ---------------------------------------------------------------------------
__device__ __forceinline__ float qlstm_wval(int k, int c,
                                            const float* Wf, const float* Wi,
                                            const float* Wu, const float* Wo,
                                            const float* Wi2k) {
  int g = c >> 3, n = c & 7;
  if (c < 32) {
    const float* W = (g == 0) ? Wf : (g == 1) ? Wi : (g == 2) ? Wu : Wo;
    return W[k * NH + n];               // W is [72][8] row-major; rows 0..63 = x part
  }
  if (c < 40) return Wi2k[k * NH + n];
  return 0.0f;
}

__global__ void qlstm_pack(const float* __restrict__ Wf, const float* __restrict__ bf,
                           const float* __restrict__ Wi, const float* __restrict__ bi,
                           const float* __restrict__ Wu, const float* __restrict__ bu,
                           const float* __restrict__ Wo, const float* __restrict__ bo,
                           const float* __restrict__ Wi2k, const float* __restrict__ bi2k,
                           float* __restrict__ Bfrag, float* __restrict__ bpack) {
  int idx = blockIdx.x * blockDim.x + threadIdx.x;      // 0 .. 16*3*32-1
  if (idx < 16 * 3 * 32) {
    int kstep = idx / 96;
    int rem   = idx % 96;
    int nt    = rem / 32;
    int lane  = rem % 32;
    int half  = lane >> 4, l16 = lane & 15;
    int krow  = kstep * 4 + half * 2;
    int col   = nt * 16 + l16;
    float v0 = qlstm_wval(krow,     col, Wf, Wi, Wu, Wo, Wi2k);
    float v1 = qlstm_wval(krow + 1, col, Wf, Wi, Wu, Wo, Wi2k);
    Bfrag[idx * 2]     = v0;
    Bfrag[idx * 2 + 1] = v1;
  }
  if (idx < 48) {
    int g = idx >> 3, n = idx & 7;
    float v = 0.0f;
    if (idx < 32)       v = ((g == 0) ? bf : (g == 1) ? bi : (g == 2) ? bu : bo)[n];
    else if (idx < 40)  v = bi2k[n];
    bpack[idx] = v;
  }
}

// ---------------------------------------------------------------------------
// xproj[T*B][40] = X[T*B][64] @ B[64][48] + bpack (cols 40..47 dropped).
// One wave per 64-row strip: 4 M-tiles x 3 N-tiles, K=64 as 16 chained
// V_WMMA_F32_16X16X4_F32 steps -> 192 WMMA per wave; per K-step only
// 4 A-frag b64 loads (non-temporal, X is stream-once) + 3 coalesced
// B-frag b64 loads (WGP$-resident).
// ---------------------------------------------------------------------------
__global__ void qlstm_xproj_gemm(const float* __restrict__ X,
                                 const float* __restrict__ Bfrag,
                                 const float* __restrict__ bpack,
                                 float* __restrict__ xproj) {
  const int lane = threadIdx.x & 31;
  const int wave = (blockIdx.x * blockDim.x + threadIdx.x) >> 5;
  const long row0 = (long)wave * 64;
  const int half = lane >> 4;           // 0: K=k0,k0+1   1: K=k0+2,k0+3
  const int l16  = lane & 15;

  const float* xr0 = X + (row0 +  0 + l16) * DIN + half * 2;
  const float* xr1 = X + (row0 + 16 + l16) * DIN + half * 2;
  const float* xr2 = X + (row0 + 32 + l16) * DIN + half * 2;
  const float* xr3 = X + (row0 + 48 + l16) * DIN + half * 2;
  const v2f* bfp = (const v2f*)Bfrag;

  v8f acc[4][3] = {};
#pragma unroll
  for (int ks = 0; ks < 16; ++ks) {
    v2f a0 = __builtin_nontemporal_load((const v2f*)(xr0 + ks * 4));
    v2f a1 = __builtin_nontemporal_load((const v2f*)(xr1 + ks * 4));
    v2f a2 = __builtin_nontemporal_load((const v2f*)(xr2 + ks * 4));
    v2f a3 = __builtin_nontemporal_load((const v2f*)(xr3 + ks * 4));
    v2f b0 = bfp[(ks * 3 + 0) * 32 + lane];
    v2f b1 = bfp[(ks * 3 + 1) * 32 + lane];
    v2f b2 = bfp[(ks * 3 + 2) * 32 + lane];
    acc[0][0] = __builtin_amdgcn_wmma_f32_16x16x4_f32(false, a0, false, b0, (short)0, acc[0][0], false, false);
    acc[0][1] = __builtin_amdgcn_wmma_f32_16x16x4_f32(false, a0, false, b1, (short)0, acc[0][1], false, false);
    acc[0][2] = __builtin_amdgcn_wmma_f32_16x16x4_f32(false, a0, false, b2, (short)0, acc[0][2], false, false);
    acc[1][0] = __builtin_amdgcn_wmma_f32_16x16x4_f32(false, a1, false, b0, (short)0, acc[1][0], false, false);
    acc[1][1] = __builtin_amdgcn_wmma_f32_16x16x4_f32(false, a1, false, b1, (short)0, acc[1][1], false, false);
    acc[1][2] = __builtin_amdgcn_wmma_f32_16x16x4_f32(false, a1, false, b2, (short)0, acc[1][2], false, false);
    acc[2][0] = __builtin_amdgcn_wmma_f32_16x16x4_f32(false, a2, false, b0, (short)0, acc[2][0], false, false);
    acc[2][1] = __builtin_amdgcn_wmma_f32_16x16x4_f32(false, a2, false, b1, (short)0, acc[2][1], false, false);
    acc[2][2] = __builtin_amdgcn_wmma_f32_16x16x4_f32(false, a2, false, b2, (short)0, acc[2][2], false, false);
    acc[3][0] = __builtin_amdgcn_wmma_f32_16x16x4_f32(false, a3, false, b0, (short)0, acc[3][0], false, false);
    acc[3][1] = __builtin_amdgcn_wmma_f32_16x16x4_f32(false, a3, false, b1, (short)0, acc[3][1], false, false);
    acc[3][2] = __builtin_amdgcn_wmma_f32_16x16x4_f32(false, a3, false, b2, (short)0, acc[3][2], false, false);
  }

  // C/D layout: VGPR r holds M = r + 8*half, N = l16 (per N-tile). Bias per-N.
  const float bia0 = bpack[l16];
  const float bia1 = bpack[16 + l16];
  const float bia2 = bpack[32 + l16];
  const int mbase = half * 8;
#pragma unroll
  for (int tile = 0; tile < 4; ++tile) {
#pragma unroll
    for (int r = 0; r < 8; ++r) {
      float* out = xproj + (row0 + tile * 16 + mbase + r) * 40;
      out[l16]      = acc[tile][0][r] + bia0;
      out[16 + l16] = acc[tile][1][r] + bia1;
      if (l16 < 8) out[32 + l16] = acc[tile][2][r] + bia2;   // cols 32..39 only
    }
  }
}

// ---------------------------------------------------------------------------
// Serial scan over T. 8 lanes per batch element (groups of 8 inside a
// wave32). Lane n owns hidden unit n; hx broadcast via __shfl, qlayer
// cumprod via shuffle prefix-scan, kernel weight via xor-shuffle product.
// xproj reads should hit L2 (84 MB resident, X was loaded NT).
// ---------------------------------------------------------------------------
__device__ __forceinline__ float qlayer_elem(float a, int n) {
  float z = __cosf(a);
  float cp = z;
  float u;
  u = __shfl_up(cp, 1, 8); if (n >= 1) cp *= u;
  u = __shfl_up(cp, 2, 8); if (n >= 2) cp *= u;
  u = __shfl_up(cp, 4, 8); if (n >= 4) cp *= u;
  float tot = __shfl(cp, 7, 8);        // product of all 8
  return (n == 0) ? (tot / z) : cp;    // z0 = prod_{i>=1} cos(a_i)
}

__device__ __forceinline__ float sigmoidf_(float x) {
  return 1.0f / (1.0f + __expf(-x));
}

__global__ void qlstm_recurrent(const float* __restrict__ xproj,
                                const float* __restrict__ Wf, const float* __restrict__ Wi,
                                const float* __restrict__ Wu, const float* __restrict__ Wo,
                                const float* __restrict__ thf, const float* __restrict__ thi,
                                const float* __restrict__ thu, const float* __restrict__ tho,
                                const float* __restrict__ Wh2k, const float* __restrict__ bh2k,
                                float* __restrict__ out) {
  const int gid = blockIdx.x * blockDim.x + threadIdx.x;   // 0 .. B*8-1
  const int b = gid >> 3;
  const int n = gid & 7;

  // Preload recurrent (h-part) weight columns: rows 64..71 of each gate W,
  // plus Wh2k column n. 40 registers, constant across the whole scan.
  float wf[8], wi[8], wu[8], wo[8], wh[8];
#pragma unroll
  for (int k = 0; k < 8; ++k) {
    wf[k] = Wf[(64 + k) * NH + n];
    wi[k] = Wi[(64 + k) * NH + n];
    wu[k] = Wu[(64 + k) * NH + n];
    wo[k] = Wo[(64 + k) * NH + n];
    wh[k] = Wh2k[k * NH + n];
  }
  const float tf = thf[n], ti = thi[n], tu = thu[n], to_ = tho[n], bh = bh2k[n];

  float hx = 0.0f, cx = 0.0f;

  for (int t = 0; t < TT; ++t) {
    const float* xp = xproj + ((long)t * BB + b) * 40;
    float sf = xp[n], si = xp[8 + n], su = xp[16 + n], so = xp[24 + n];
    float ky = xp[32 + n];
    float kx = bh;
#pragma unroll
    for (int k = 0; k < 8; ++k) {
      float hk = __shfl(hx, k, 8);
      sf += hk * wf[k];
      si += hk * wi[k];
      su += hk * wu[k];
      so += hk * wo[k];
      kx += hk * wh[k];
    }

    // kernel weight w = | prod_n cos(0.5*(kx - ky)) |, broadcast over group
    float kc = __cosf(0.5f * (kx - ky));
    kc *= __shfl_xor(kc, 1, 8);
    kc *= __shfl_xor(kc, 2, 8);
    kc *= __shfl_xor(kc, 4, 8);
    float w = fabsf(kc);

    float f  = sigmoidf_(qlayer_elem(sf + tf, n) * w);
    float iv = sigmoidf_(qlayer_elem(si + ti, n) * w);
    float g  = tanhf(qlayer_elem(su + tu, n) * w);
    float o  = sigmoidf_(qlayer_elem(so + to_, n) * w);

    cx = f * cx + iv * g;
    hx = o * tanhf(cx);

    out[((long)t * BB + b) * NH + n] = hx;
  }
  // final hx, cx appended after outputs
  out[(long)TT * BB * NH + (long)b * NH + n] = hx;
  out[(long)TT * BB * NH + (long)BB * NH + (long)b * NH + n] = cx;
}

// ---------------------------------------------------------------------------
extern "C" void kernel_launch(void* const* d_in, const int* in_sizes, int n_in,
                              void* d_out, int out_size, void* d_ws, size_t ws_size,
                              hipStream_t stream) {
  // setup_inputs() dict order:
  // 0 inputs, 1 Wf, 2 bf, 3 th_f, 4 Wi, 5 bi, 6 th_i, 7 Wu, 8 bu, 9 th_u,
  // 10 Wo, 11 bo, 12 th_o, 13 Wh2k, 14 bh2k, 15 Wi2k, 16 bi2k
  const float* X    = (const float*)d_in[0];
  const float* Wf   = (const float*)d_in[1];
  const float* bf   = (const float*)d_in[2];
  const float* thf  = (const float*)d_in[3];
  const float* Wi   = (const float*)d_in[4];
  const float* bi   = (const float*)d_in[5];
  const float* thi  = (const float*)d_in[6];
  const float* Wu   = (const float*)d_in[7];
  const float* bu   = (const float*)d_in[8];
  const float* thu  = (const float*)d_in[9];
  const float* Wo   = (const float*)d_in[10];
  const float* bo   = (const float*)d_in[11];
  const float* tho  = (const float*)d_in[12];
  const float* Wh2k = (const float*)d_in[13];
  const float* bh2k = (const float*)d_in[14];
  const float* Wi2k = (const float*)d_in[15];
  const float* bi2k = (const float*)d_in[16];

  float* ws    = (float*)d_ws;
  float* Bfrag = ws;               // 16*3*32*2 = 3072 floats (fragment layout)
  float* bpack = ws + 3072;        // 48 floats
  float* xproj = ws + 4096;        // T*B*40 floats = 83.9 MB

  qlstm_pack<<<dim3(6), dim3(256), 0, stream>>>(Wf, bf, Wi, bi, Wu, bu, Wo, bo,
                                                Wi2k, bi2k, Bfrag, bpack);

  // M = T*B = 524288 rows -> 8192 waves (one 64-row strip each), 8 waves/block
  qlstm_xproj_gemm<<<dim3(1024), dim3(256), 0, stream>>>(X, Bfrag, bpack, xproj);

  // B*8 = 8192 threads, 8 lanes per batch element
  qlstm_recurrent<<<dim3(32), dim3(256), 0, stream>>>(xproj, Wf, Wi, Wu, Wo,
                                                      thf, thi, thu, tho,
                                                      Wh2k, bh2k, (float*)d_out);
}